// GenerativeContrastiveModelling_23699629540092
// MI455X (gfx1250) — compile-verified
//
#include <hip/hip_runtime.h>
#include <hip/hip_bf16.h>
#include <cstddef>
#include <cstdint>

// Problem constants (fixed by the reference harness).
#define GCM_B 8
#define GCM_N 65536
#define GCM_E 64
#define GCM_C 128
#define GCM_CHUNK 2048          // examples per block
#define GCM_WAVES 8             // 256 threads / wave32

// ---------------------------------------------------------------------------
// Kernel 0: zero the global workspace accumulators.
// ---------------------------------------------------------------------------
__global__ void gcm_zero_kernel(float* __restrict__ ws, int n) {
    int i = blockIdx.x * blockDim.x + threadIdx.x;
    if (i < n) ws[i] = 0.0f;
}

// ---------------------------------------------------------------------------
// Kernel 1: LDS-privatized segmented accumulation.
//   LDS: sumP[C][E], sumPM[C][E], s[C], cnt[C]  (~66.5 KB) + targets[2048] (8 KB).
//   Block start: issue async global->LDS copies of the chunk's targets
//   (global_load_async_to_lds_b128, ASYNCcnt) so the fetch overlaps the LDS
//   accumulator zeroing; s_wait_asynccnt + barrier publishes them.
//   Per example: one wave loads 64 elements as float2 (coalesced b64),
//   computes p, p*m and the scalar (log p - p*m^2), wave-reduces the scalar
//   with shfl_xor (wave32), then 4 ds_add_f32 per lane + 2 on lane 0.
//   Finally flushes LDS partials with global_atomic_add_f32.
// ---------------------------------------------------------------------------
__global__ __launch_bounds__(256) void gcm_accum_kernel(
    const float* __restrict__ means,
    const float* __restrict__ precisions,
    const int*   __restrict__ targets,
    float* __restrict__ wsSumP,    // [B][C][E]
    float* __restrict__ wsSumPM,   // [B][C][E]
    float* __restrict__ wsS,       // [B][C]
    float* __restrict__ wsCnt)     // [B][C]
{
    extern __shared__ float smem[];
    float* sP   = smem;                          // C*E floats
    float* sPM  = smem + GCM_C * GCM_E;          // C*E floats
    float* sS   = smem + 2 * GCM_C * GCM_E;      // C floats
    float* sCnt = sS + GCM_C;                    // C floats
    int*   sTgt = (int*)(smem + 2 * GCM_C * GCM_E + 2 * GCM_C);  // CHUNK ints

    const int tid = threadIdx.x;

    const int blocksPerBatch = GCM_N / GCM_CHUNK;
    const int b  = blockIdx.x / blocksPerBatch;
    const int n0 = (blockIdx.x % blocksPerBatch) * GCM_CHUNK;
    const int* tg = targets + (size_t)b * GCM_N + n0;

    // --- Stage chunk targets into LDS asynchronously (16 B per lane-slot).
    // 2048 ints = 512 x b128 slots; 256 threads -> 2 issues each.
    for (int s = tid; s < GCM_CHUNK / 4; s += blockDim.x) {
        uint64_t gaddr = (uint64_t)(uintptr_t)(tg + s * 4);
        uint32_t laddr = (uint32_t)(uintptr_t)(sTgt + s * 4);  // LDS offset = addr[31:0]
        asm volatile("global_load_async_to_lds_b128 %0, %1, off"
                     :: "v"(laddr), "v"(gaddr) : "memory");
    }

    // --- Zero LDS accumulators (overlaps the async target fetch).
    const int nLds = 2 * GCM_C * GCM_E + 2 * GCM_C;
    for (int i = tid; i < nLds; i += blockDim.x) smem[i] = 0.0f;

    // Wait for this wave's async copies, then publish to the workgroup.
    asm volatile("s_wait_asynccnt 0x0" ::: "memory");
    __syncthreads();

    const int wave = tid >> 5;
    const int lane = tid & 31;

    const size_t base = ((size_t)b * GCM_N + (size_t)n0) * GCM_E;
    const float2* m2 = reinterpret_cast<const float2*>(means + base);
    const float2* p2 = reinterpret_cast<const float2*>(precisions + base);

    // Each wave handles examples i = wave, wave+8, ... within the chunk.
    #pragma unroll 2
    for (int i = wave; i < GCM_CHUNK; i += GCM_WAVES) {
        // Prefetch 32 examples ahead (global_prefetch_b8 on gfx1250).
        const int ip = i + 4 * GCM_WAVES;
        if (ip < GCM_CHUNK) {
            __builtin_prefetch(&m2[(size_t)ip * 32 + lane], 0, 0);
            __builtin_prefetch(&p2[(size_t)ip * 32 + lane], 0, 0);
        }

        const float2 m = m2[(size_t)i * 32 + lane];   // e = 2*lane, 2*lane+1
        const float2 p = p2[(size_t)i * 32 + lane];
        const int t = sTgt[i];                        // wave-uniform LDS read

        const float pmx = p.x * m.x;
        const float pmy = p.y * m.y;

        // Scalar contribution: log p - p*m^2 (only its E-sum is ever needed).
        float sloc = __logf(p.x) + __logf(p.y) - pmx * m.x - pmy * m.y;
        #pragma unroll
        for (int off = 16; off > 0; off >>= 1)
            sloc += __shfl_xor(sloc, off, 32);

        const int e = lane * 2;
        atomicAdd(&sP [t * GCM_E + e    ], p.x);
        atomicAdd(&sP [t * GCM_E + e + 1], p.y);
        atomicAdd(&sPM[t * GCM_E + e    ], pmx);
        atomicAdd(&sPM[t * GCM_E + e + 1], pmy);
        if (lane == 0) {
            atomicAdd(&sS[t], sloc);
            atomicAdd(&sCnt[t], 1.0f);
        }
    }
    __syncthreads();

    // Flush LDS partials to the global workspace.
    const size_t gce = (size_t)b * GCM_C * GCM_E;
    for (int i = tid; i < GCM_C * GCM_E; i += blockDim.x) {
        atomicAdd(&wsSumP [gce + i], sP[i]);
        atomicAdd(&wsSumPM[gce + i], sPM[i]);
    }
    const size_t gc = (size_t)b * GCM_C;
    for (int i = tid; i < GCM_C; i += blockDim.x) {
        atomicAdd(&wsS  [gc + i], sS[i]);
        atomicAdd(&wsCnt[gc + i], sCnt[i]);
    }
}

// ---------------------------------------------------------------------------
// Kernel 2: finalize. One wave32 per (b,c) pair; lane handles e=2l,2l+1.
//   product_mean = sumPM / sumP ; product_precision = sumP
//   log_norm = 0.5*(1-ns)*E*log(2pi)
//            + 0.5*( sum_e[sumP*pm^2 - log(sumP)] + s )
// ---------------------------------------------------------------------------
__global__ __launch_bounds__(256) void gcm_final_kernel(
    const float* __restrict__ wsSumP,
    const float* __restrict__ wsSumPM,
    const float* __restrict__ wsS,
    const float* __restrict__ wsCnt,
    float* __restrict__ out)
{
    const int tid  = threadIdx.x;
    const int wave = tid >> 5;
    const int lane = tid & 31;
    const int pair = blockIdx.x * GCM_WAVES + wave;   // b*C + c
    if (pair >= GCM_B * GCM_C) return;

    const size_t off = (size_t)pair * GCM_E;
    const float2 sp  = reinterpret_cast<const float2*>(wsSumP  + off)[lane];
    const float2 spm = reinterpret_cast<const float2*>(wsSumPM + off)[lane];

    float2 pm;
    pm.x = spm.x / sp.x;
    pm.y = spm.y / sp.y;

    float* outMean = out;
    float* outPrec = out + (size_t)GCM_B * GCM_C * GCM_E;
    float* outLog  = out + (size_t)2 * GCM_B * GCM_C * GCM_E;

    reinterpret_cast<float2*>(outMean + off)[lane] = pm;
    reinterpret_cast<float2*>(outPrec + off)[lane] = sp;

    float term = sp.x * pm.x * pm.x - __logf(sp.x)
               + sp.y * pm.y * pm.y - __logf(sp.y);
    #pragma unroll
    for (int o = 16; o > 0; o >>= 1)
        term += __shfl_xor(term, o, 32);

    if (lane == 0) {
        float ns = wsCnt[pair];
        if (ns < 1.0f) ns = 1.0f;
        const float log2pi = 1.8378770664093453f;  // ln(2*pi)
        outLog[pair] = 0.5f * (1.0f - ns) * (float)GCM_E * log2pi
                     + 0.5f * (term + wsS[pair]);
    }
}

// ---------------------------------------------------------------------------
// Launch: zero ws -> accumulate -> finalize (stream-ordered).
// ---------------------------------------------------------------------------
extern "C" void kernel_launch(void* const* d_in, const int* in_sizes, int n_in,
                              void* d_out, int out_size, void* d_ws, size_t ws_size,
                              hipStream_t stream) {
    (void)in_sizes; (void)n_in; (void)out_size; (void)ws_size;

    const float* means      = (const float*)d_in[0];
    const float* precisions = (const float*)d_in[1];
    const int*   targets    = (const int*)d_in[2];
    // d_in[3] = num_classes scalar; fixed at 128 by the reference setup.

    float* ws      = (float*)d_ws;
    float* wsSumP  = ws;
    float* wsSumPM = ws + (size_t)GCM_B * GCM_C * GCM_E;
    float* wsS     = ws + (size_t)2 * GCM_B * GCM_C * GCM_E;
    float* wsCnt   = wsS + (size_t)GCM_B * GCM_C;

    const int nWs = 2 * GCM_B * GCM_C * GCM_E + 2 * GCM_B * GCM_C;
    gcm_zero_kernel<<<(nWs + 255) / 256, 256, 0, stream>>>(ws, nWs);

    const size_t smemBytes =
        (size_t)(2 * GCM_C * GCM_E + 2 * GCM_C) * sizeof(float)   // accumulators
        + (size_t)GCM_CHUNK * sizeof(int);                        // staged targets
    const int nBlocks = GCM_B * (GCM_N / GCM_CHUNK);              // 256 blocks
    gcm_accum_kernel<<<nBlocks, 256, smemBytes, stream>>>(
        means, precisions, targets, wsSumP, wsSumPM, wsS, wsCnt);

    gcm_final_kernel<<<(GCM_B * GCM_C) / GCM_WAVES, 256, 0, stream>>>(
        wsSumP, wsSumPM, wsS, wsCnt, (float*)d_out);
}